// LoRALiMEEmbedding_17325898072233
// MI455X (gfx1250) — compile-verified
//
#include <hip/hip_runtime.h>
#include <stdint.h>

// ---------------- problem constants (from reference) ----------------
#define H            2048
#define RANK         16
#define NEXP         8
#define TOKENS       8192            // BATCH(4) * SEQ(2048)
#define SCALING      1.0f            // ALPHA/RANK = 16/16
#define GAMMA_ROUT   0.5f
#define INV_SOFT_T   2.0f            // 1/SOFT_TOPK_TEMP
#define EPSF         1e-6f

typedef __attribute__((ext_vector_type(2))) float v2f;
typedef __attribute__((ext_vector_type(8))) float v8f;

// ---------------- K0: zero the two abs-max accumulators ----------------
__global__ void k0_init(unsigned* scales) {
    if (threadIdx.x < 2) scales[threadIdx.x] = 0u;
}

// ---------------- K1: delta_r = E_out @ A^T  (WMMA f32 16x16x4) --------
// One block = one 16-token tile. 8 waves, wave w owns K-slice [256w, 256w+256).
// Also produces E_sl, d_sl and global abs-max scales.
__global__ __launch_bounds__(256) void k1_lora_down(
    const int* __restrict__ ids, const float* __restrict__ emb,
    const float* __restrict__ Amat,   // (RANK, H) row-major
    const float* __restrict__ Blora,  // (H, RANK) row-major
    float* __restrict__ delta_r, float* __restrict__ e_sl,
    float* __restrict__ d_sl, unsigned* __restrict__ scales)
{
    __shared__ int   rows[16];
    __shared__ float tile[16][18];   // padded: conflict-free swizzled access
    __shared__ float red[8][2];

    const int tid  = threadIdx.x;
    const int wave = tid >> 5;
    const int lane = tid & 31;
    const int tb   = blockIdx.x * 16;

    if (tid < 16) rows[tid] = ids[tb + tid];
    { int t = tid >> 4, r = tid & 15; tile[t][r] = 0.0f; }
    __syncthreads();

    // A-fragment swizzle (ISA 16x4 f32 A layout):
    //   lanes 0-15 : M=lane,    VGPR0=K0,  VGPR1=K1
    //   lanes 16-31: M=lane-16, VGPR0=K2,  VGPR1=K3
    // B-fragment (4x16): VGPRv lanes0-15 row k=v, lanes16-31 row k=v+2.
    const int  mrow  = lane & 15;
    const int  khi   = (lane >= 16) ? 2 : 0;
    const long ebase = (long)rows[mrow] * H;
    const int  kbase = wave * 256;

    v8f c = {0.f,0.f,0.f,0.f,0.f,0.f,0.f,0.f};
    for (int j = 0; j < 64; ++j) {
        const int k0 = kbase + j * 4 + khi;
        v2f a = *(const v2f*)(emb  + ebase + k0);          // E_out[tok][k0,k0+1]
        v2f b = *(const v2f*)(Amat + (long)mrow * H + k0); // A[r=n][k0,k0+1]
        c = __builtin_amdgcn_wmma_f32_16x16x4_f32(
                false, a, false, b, (short)0, c, false, false);
    }

    // reduce 8 partial 16(tok)x16(rank) tiles across waves via LDS float atomics
    #pragma unroll
    for (int v = 0; v < 8; ++v) {
        const int m = v + ((lane >= 16) ? 8 : 0);   // C layout: VGPRv -> M=v / v+8
        atomicAdd(&tile[m][mrow], c[v]);
    }
    __syncthreads();

    { // write delta_r tile (coalesced)
        int t = tid >> 4, r = tid & 15;
        delta_r[(long)(tb + t) * RANK + r] = tile[t][r];
    }

    // E_sl / d_sl for these 16 tokens (threads 0..127 = waves 0..3)
    float lv_e = 0.0f, lv_d = 0.0f;
    if (tid < 128) {
        int t = tid >> 3, e = tid & 7;
        float ev = emb[(long)rows[t] * H + e];
        float dv = 0.0f;
        #pragma unroll
        for (int r = 0; r < RANK; ++r) dv += tile[t][r] * Blora[e * RANK + r];
        dv *= SCALING;
        e_sl[(tb + t) * NEXP + e] = ev;
        d_sl[(tb + t) * NEXP + e] = dv;
        lv_e = fabsf(ev);
        lv_d = fabsf(dv);
    }
    for (int off = 16; off; off >>= 1) {
        lv_e = fmaxf(lv_e, __shfl_xor(lv_e, off, 32));
        lv_d = fmaxf(lv_d, __shfl_xor(lv_d, off, 32));
    }
    if (lane == 0) { red[wave][0] = lv_e; red[wave][1] = lv_d; }
    __syncthreads();
    if (tid == 0) {
        float me = 0.0f, md = 0.0f;
        for (int w = 0; w < 8; ++w) { me = fmaxf(me, red[w][0]); md = fmaxf(md, red[w][1]); }
        // abs values are >= 0, so IEEE-bit uint compare == float compare
        atomicMax(&scales[0], __float_as_uint(me));
        atomicMax(&scales[1], __float_as_uint(md));
    }
}

// ---------------- K1b: per-token routing weights -----------------------
__global__ __launch_bounds__(256) void k1b_route(
    const float* __restrict__ e_sl, const float* __restrict__ d_sl,
    const unsigned* __restrict__ scales, float* __restrict__ wts)
{
    const int t = blockIdx.x * blockDim.x + threadIdx.x;
    if (t >= TOKENS) return;
    const float hs = fmaxf(__uint_as_float(scales[0]), EPSF);
    const float ds = fmaxf(__uint_as_float(scales[1]), EPSF);

    float p[NEXP];
    float m = -1e30f;
    #pragma unroll
    for (int e = 0; e < NEXP; ++e) {
        float v = (1.0f - GAMMA_ROUT) * (e_sl[t * NEXP + e] / hs)
                + GAMMA_ROUT         * (d_sl[t * NEXP + e] / ds);
        p[e] = v; m = fmaxf(m, v);
    }
    float s = 0.0f;
    #pragma unroll
    for (int e = 0; e < NEXP; ++e) { p[e] = __expf(p[e] - m); s += p[e]; }
    const float inv = 1.0f / s;
    #pragma unroll
    for (int e = 0; e < NEXP; ++e) p[e] *= inv;      // probs

    float m1 = -1.0f, m2 = -1.0f;                    // top-2 -> threshold = 2nd
    #pragma unroll
    for (int e = 0; e < NEXP; ++e) {
        float v = p[e];
        if (v > m1) { m2 = m1; m1 = v; } else if (v > m2) { m2 = v; }
    }
    float sum = 0.0f, w[NEXP];
    #pragma unroll
    for (int e = 0; e < NEXP; ++e) {
        float mask = 1.0f / (1.0f + __expf(-(p[e] - m2) * INV_SOFT_T));
        w[e] = p[e] * mask; sum += w[e];
    }
    const float invs = 1.0f / (sum + 1e-9f);
    #pragma unroll
    for (int e = 0; e < NEXP; ++e) wts[t * NEXP + e] = w[e] * invs;
}

// ---------------- K2: fused delta-up GEMM + p_mix + output -------------
// Block = one 16-token tile (shares ids/delta_r/weights in LDS across 8 waves);
// wave (blockIdx.y*8 + wave) owns one 16-wide h tile: 4x WMMA f32 16x16x4.
__global__ __launch_bounds__(256) void k2_fused_out(
    const int* __restrict__ ids, const float* __restrict__ emb,
    const float* __restrict__ Blora, const float* __restrict__ limes,
    const float* __restrict__ lshared, const float* __restrict__ gamma,
    const float* __restrict__ delta_r, const float* __restrict__ wts,
    float* __restrict__ out)
{
    __shared__ int   rows[16];
    __shared__ float dr[16][18];      // delta_r tile, padded rows
    __shared__ float wsh[16][NEXP];   // routing weights tile

    const int tid  = threadIdx.x;
    const int wave = tid >> 5;
    const int lane = tid & 31;
    const int tb   = blockIdx.x * 16;

    if (tid < 16) rows[tid] = ids[tb + tid];
    { int t = tid >> 4, r = tid & 15; dr[t][r] = delta_r[(long)(tb + t) * RANK + r]; }
    if (tid < 128) { int t = tid >> 3, e = tid & 7; wsh[t][e] = wts[(tb + t) * NEXP + e]; }
    __syncthreads();

    const int hb   = (blockIdx.y * 8 + wave) * 16;
    const int n    = lane & 15;            // this lane's column (h within tile)
    const int h    = hb + n;
    const int khi  = (lane >= 16) ? 2 : 0;

    // D = delta_r(16x16) x B_lora^T(16x16) over K=RANK in 4 steps
    v8f c = {0.f,0.f,0.f,0.f,0.f,0.f,0.f,0.f};
    #pragma unroll
    for (int j = 0; j < 4; ++j) {
        const int k0 = j * 4 + khi;
        v2f a = *(const v2f*)(&dr[n][k0]);                     // A frag from LDS
        v2f b = *(const v2f*)(Blora + (long)h * RANK + k0);    // B[k][n]=B_lora[h][k]
        c = __builtin_amdgcn_wmma_f32_16x16x4_f32(
                false, a, false, b, (short)0, c, false, false);
    }

    // per-lane column data for p_mix (LiMEs is 64KB -> L2 resident)
    float pe[NEXP];
    #pragma unroll
    for (int e = 0; e < NEXP; ++e) pe[e] = limes[e * H + h];
    const float shv = lshared[h];
    const float g   = 1.0f / (1.0f + __expf(-gamma[0]));

    #pragma unroll
    for (int v = 0; v < 8; ++v) {
        const int t = v + ((lane >= 16) ? 8 : 0);
        float pm = 0.0f;
        #pragma unroll
        for (int e = 0; e < NEXP; ++e) pm += wsh[t][e] * pe[e];
        pm = (1.0f - g) * pm + g * shv;
        const float eo = emb[(long)rows[t] * H + h];   // L2 hit after K1
        out[(long)(tb + t) * H + h] = eo + SCALING * c[v] * pm;
    }
}

// ---------------- launcher ---------------------------------------------
extern "C" void kernel_launch(void* const* d_in, const int* in_sizes, int n_in,
                              void* d_out, int out_size, void* d_ws, size_t ws_size,
                              hipStream_t stream) {
    const int*   ids     = (const int*)  d_in[0];
    const float* emb     = (const float*)d_in[1];
    const float* Amat    = (const float*)d_in[2];
    const float* Blora   = (const float*)d_in[3];
    const float* limes   = (const float*)d_in[4];
    const float* lshared = (const float*)d_in[5];
    const float* gamma   = (const float*)d_in[6];
    float*       out     = (float*)d_out;

    // workspace layout (~1.3 MB): [scales u32x2 | delta_r | E_sl | d_sl | weights]
    char*     ws      = (char*)d_ws;
    unsigned* scales  = (unsigned*)ws;
    float*    delta_r = (float*)(ws + 16);
    float*    e_sl    = delta_r + (size_t)TOKENS * RANK;
    float*    d_sl    = e_sl    + (size_t)TOKENS * NEXP;
    float*    wts     = d_sl    + (size_t)TOKENS * NEXP;

    hipLaunchKernelGGL(k0_init, dim3(1), dim3(32), 0, stream, scales);
    hipLaunchKernelGGL(k1_lora_down, dim3(TOKENS / 16), dim3(256), 0, stream,
                       ids, emb, Amat, Blora, delta_r, e_sl, d_sl, scales);
    hipLaunchKernelGGL(k1b_route, dim3(TOKENS / 256), dim3(256), 0, stream,
                       e_sl, d_sl, scales, wts);
    hipLaunchKernelGGL(k2_fused_out, dim3(TOKENS / 16, H / 128), dim3(256), 0, stream,
                       ids, emb, Blora, limes, lshared, gamma, delta_r, wts, out);
}